// _AnchorTargetLayer_xy_74122545595033
// MI455X (gfx1250) — compile-verified
//
#include <hip/hip_runtime.h>
#include <stdint.h>
#include <math.h>

#define A_N 210000
#define B_N 8
#define G_N 15
#define GTS 7            // gt_tubes row stride
#define NUM_FG 128
#define RPN_BS 256
#define NEG_OV 0.3f
#define POS_OV 0.7f
#define HUNROLL 8

// ---------------- workspace layout (bytes) ----------------
#define WS_ARGMAX 0                                          // u8 [B*A]
#define WS_LABELS (WS_ARGMAX + ((B_N*A_N + 255) & ~255))     // i8 [B*A]
#define WS_GTENC  (WS_LABELS + ((B_N*A_N + 255) & ~255))     // u32[B*G]
#define WS_GTADJ  (WS_GTENC + 512)                           // f32[B*G]
#define WS_CNTFG  (WS_GTADJ + 512)                           // u32[8]
#define WS_CNTBG  (WS_CNTFG + 64)                            // u32[8]
#define WS_PREFIX (WS_CNTBG + 64)                            // u32[16]
#define WS_KREM   (WS_PREFIX + 64)                           // u32[16]
#define WS_ACTIVE (WS_KREM + 64)                             // u32[16]
#define WS_THR    (WS_ACTIVE + 64)                           // u32[16]
#define WS_UW     (WS_THR + 64)                              // f32[1]
#define WS_HIST   (WS_UW + 64)                               // u32[16*256]

// ---------------- CDNA5 async global->LDS staging ----------------
typedef __attribute__((address_space(1))) int as1_int;
typedef __attribute__((address_space(3))) int as3_int;

#if __has_builtin(__builtin_amdgcn_global_load_async_to_lds_b32)
#define HAVE_ASYNC_LDS 1
#else
#define HAVE_ASYNC_LDS 0
#endif

__device__ __forceinline__ void wait_async0() {
#if __has_builtin(__builtin_amdgcn_s_wait_asynccnt)
  __builtin_amdgcn_s_wait_asynccnt(0);
#elif defined(__gfx1250__)
  asm volatile("s_wait_asynccnt 0" ::: "memory");
#endif
}

__device__ __forceinline__ void async_dword(const float* gsrc, float* sdst) {
#if HAVE_ASYNC_LDS
  as1_int* g = (as1_int*)(uintptr_t)gsrc;
  as3_int* l = (as3_int*)(uint32_t)(uintptr_t)sdst;
  __builtin_amdgcn_global_load_async_to_lds_b32(g, l, 0, 0);
#else
  *sdst = *gsrc;
#endif
}

// copy n dwords global->LDS; caller must __syncthreads() afterwards
__device__ __forceinline__ void stage_dwords(const float* __restrict__ gsrc,
                                             float* sdst, int tid, int n) {
  for (int i = tid; i < n; i += 256) async_dword(gsrc + i, sdst + i);
#if HAVE_ASYNC_LDS
  wait_async0();
#endif
}

// ---------------- helpers ----------------
// order-preserving float->uint encoding (works for any float; priorities >=0)
__device__ __forceinline__ unsigned fenc(float f) {
  unsigned u = __float_as_uint(f);
  return (u & 0x80000000u) ? ~u : (u | 0x80000000u);
}
__device__ __forceinline__ float fdec(unsigned e) {
  unsigned u = (e & 0x80000000u) ? (e ^ 0x80000000u) : ~e;
  return __uint_as_float(u);
}

__device__ __forceinline__ float iou3d_f(float a0, float a1, float a2,
                                         float a3, float a4, float a5,
                                         const float* g) {
  float va = (a3 - a0 + 1.f) * (a4 - a1 + 1.f) * (a5 - a2 + 1.f);
  float vg = (g[3] - g[0] + 1.f) * (g[4] - g[1] + 1.f) * (g[5] - g[2] + 1.f);
  float iw = fminf(a3, g[3]) - fmaxf(a0, g[0]) + 1.f; iw = fmaxf(iw, 0.f);
  float ih = fminf(a4, g[4]) - fmaxf(a1, g[1]) + 1.f; ih = fmaxf(ih, 0.f);
  float it = fminf(a5, g[5]) - fmaxf(a2, g[2]) + 1.f; it = fmaxf(it, 0.f);
  float inter = iw * ih * it;
  return inter / (va + vg - inter);
}

__device__ __forceinline__ void load_anc(const float* __restrict__ anchors, int a,
                                         float im0, float im1,
                                         float& a0, float& a1, float& a2,
                                         float& a3, float& a4, float& a5,
                                         bool& inside) {
  const float* ar = anchors + (size_t)a * 6;
  float x1 = ar[0], y1 = ar[1], z = ar[2], x2 = ar[3], y2 = ar[4], dur = ar[5];
  a0 = x1; a1 = y1; a2 = z + dur + 1.f;
  a3 = x2; a4 = y2; a5 = dur + dur + 1.f;
  inside = (a0 >= 0.f) && (a1 >= 0.f) && (a3 < im1) && (a4 < im0);
}

// ---------------- kernels ----------------
__global__ void k_init(unsigned int* gtenc, unsigned int* cnt_fg,
                       unsigned int* cnt_bg, unsigned int* hist) {
  int i = blockIdx.x * 256 + threadIdx.x;
  if (i < 16 * 256) hist[i] = 0u;
  if (i < B_N * G_N) gtenc[i] = 0x407FFFFFu;  // fenc(-1.0f)
  if (i < B_N) { cnt_fg[i] = 0u; cnt_bg[i] = 0u; }
}

// per (b,a): IoU row, argmax, and gt_max reduction (ordered-uint atomicMax)
__global__ void k_overlap(const float* __restrict__ anchors,
                          const float* __restrict__ gt,
                          const float* __restrict__ im_info,
                          unsigned char* __restrict__ argmax_ws,
                          unsigned int* __restrict__ gtenc) {
  __shared__ float sgt[G_N * GTS];
  __shared__ float wred[8 * G_N];
  int b = blockIdx.y;
  int tid = threadIdx.x;
  stage_dwords(gt + (size_t)b * G_N * GTS, sgt, tid, G_N * GTS);
  __syncthreads();

  int a = blockIdx.x * 256 + tid;
  bool valid = a < A_N;
  float im0 = im_info[0], im1 = im_info[1];
  float a0 = 0, a1 = 0, a2 = 0, a3 = 0, a4 = 0, a5 = 0;
  bool inside = false;
  if (valid) load_anc(anchors, a, im0, im1, a0, a1, a2, a3, a4, a5, inside);

  float mv = -1e30f; int idx = 0;
  for (int g = 0; g < G_N; ++g) {
    float v = iou3d_f(a0, a1, a2, a3, a4, a5, &sgt[g * GTS]);
    if (valid && v > mv) { mv = v; idx = g; }
    float r = (valid && inside) ? v : -1.f;
    // wave32 max reduce
    for (int o = 16; o > 0; o >>= 1) r = fmaxf(r, __shfl_xor(r, o, 32));
    if ((tid & 31) == 0) wred[(tid >> 5) * G_N + g] = r;
  }
  __syncthreads();
  if (tid < G_N) {
    float m = wred[tid];
    for (int w = 1; w < 8; ++w) m = fmaxf(m, wred[w * G_N + tid]);
    atomicMax(&gtenc[b * G_N + tid], fenc(m));
  }
  if (valid) argmax_ws[(size_t)b * A_N + a] = (unsigned char)idx;
}

__global__ void k_decode(const unsigned int* __restrict__ gtenc,
                         float* __restrict__ gtadj) {
  int i = threadIdx.x;
  if (i < B_N * G_N) {
    float f = fdec(gtenc[i]);
    if (f == 0.f) f = 1e-5f;
    gtadj[i] = f;
  }
}

// per (b,a): pre-subsample labels; per-batch fg/bg counts
__global__ void k_labels(const float* __restrict__ anchors,
                         const float* __restrict__ gt,
                         const float* __restrict__ im_info,
                         const float* __restrict__ gtadj,
                         signed char* __restrict__ labels_ws,
                         unsigned int* __restrict__ cnt_fg,
                         unsigned int* __restrict__ cnt_bg) {
  __shared__ float sgt[G_N * GTS];
  __shared__ float sadj[G_N];
  __shared__ unsigned int c01[2];
  int b = blockIdx.y;
  int tid = threadIdx.x;
  if (tid == 0) { c01[0] = 0u; c01[1] = 0u; }
  if (tid < G_N * GTS) async_dword(gt + (size_t)b * G_N * GTS + tid, &sgt[tid]);
  if (tid < G_N) async_dword(gtadj + b * G_N + tid, &sadj[tid]);
#if HAVE_ASYNC_LDS
  wait_async0();
#endif
  __syncthreads();

  int a = blockIdx.x * 256 + tid;
  if (a < A_N) {
    float im0 = im_info[0], im1 = im_info[1];
    float a0, a1, a2, a3, a4, a5; bool inside;
    load_anc(anchors, a, im0, im1, a0, a1, a2, a3, a4, a5, inside);
    float mv = -1e30f; bool best = false;
    for (int g = 0; g < G_N; ++g) {
      float v = iou3d_f(a0, a1, a2, a3, a4, a5, &sgt[g * GTS]);
      if (v > mv) mv = v;
      if (v == sadj[g]) best = true;
    }
    int lab = -1;
    if (inside) {
      if (mv < NEG_OV) lab = 0;
      if (best) lab = 1;
      if (mv >= POS_OV) lab = 1;
    }
    labels_ws[(size_t)b * A_N + a] = (signed char)lab;
    if (lab == 1) atomicAdd(&c01[1], 1u);
    if (lab == 0) atomicAdd(&c01[0], 1u);
  }
  __syncthreads();
  if (tid == 0) {
    if (c01[1]) atomicAdd(&cnt_fg[b], c01[1]);
    if (c01[0]) atomicAdd(&cnt_bg[b], c01[0]);
  }
}

// derive selection state + uw (batch B-1)
__global__ void k_setup(const unsigned int* __restrict__ cnt_fg,
                        const unsigned int* __restrict__ cnt_bg,
                        unsigned int* prefix, unsigned int* krem,
                        unsigned int* active, unsigned int* thr, float* uw) {
  int b = threadIdx.x;
  if (b < B_N) {
    unsigned cf = cnt_fg[b], cb = cnt_bg[b];
    int fi = b * 2, bi = b * 2 + 1;
    prefix[fi] = 0u; thr[fi] = 0xFFFFFFFFu;
    prefix[bi] = 0u; thr[bi] = 0xFFFFFFFFu;
    active[fi] = (cf > NUM_FG) ? 1u : 0u;
    krem[fi] = NUM_FG - 1;
    unsigned nfg = cf < NUM_FG ? cf : NUM_FG;
    unsigned nbg_t = RPN_BS - nfg;   // >= 128
    active[bi] = (cb > nbg_t) ? 1u : 0u;
    krem[bi] = nbg_t - 1;
    if (b == B_N - 1) {
      unsigned nbg = cb < nbg_t ? cb : nbg_t;
      uw[0] = 1.0f / (float)(nfg + nbg);
    }
  }
}

// one radix-select histogram pass (both fg and bg sets)
__global__ void k_hist(const float* __restrict__ prio,
                       const signed char* __restrict__ labels_ws,
                       const unsigned int* __restrict__ prefix,
                       const unsigned int* __restrict__ active,
                       unsigned int* __restrict__ hist, int pass) {
  __shared__ unsigned int lh[2 * 256];
  int b = blockIdx.y;
  int tid = threadIdx.x;
  lh[tid] = 0u; lh[256 + tid] = 0u;
  __syncthreads();
  int shift = 24 - 8 * pass;
  unsigned pf = prefix[b * 2], pb = prefix[b * 2 + 1];
  unsigned af = active[b * 2], ab = active[b * 2 + 1];
  int base = blockIdx.x * (256 * HUNROLL);
  for (int u = 0; u < HUNROLL; ++u) {
    int a = base + u * 256 + tid;
    if (a < A_N) {
      int lp = labels_ws[(size_t)b * A_N + a];
      bool isfg = (lp == 1) && af;
      bool isbg = (lp == 0) && ab;
      if (isfg || isbg) {
        unsigned key = fenc(prio[(size_t)b * A_N + a]);
        int s = isbg ? 1 : 0;
        unsigned p = isbg ? pb : pf;
        bool match = (pass == 0) || ((key >> (shift + 8)) == (p >> (shift + 8)));
        if (match) atomicAdd(&lh[s * 256 + ((key >> shift) & 0xFF)], 1u);
      }
    }
  }
  __syncthreads();
  unsigned h0 = lh[tid], h1 = lh[256 + tid];
  if (h0) atomicAdd(&hist[(b * 2 + 0) * 256 + tid], h0);
  if (h1) atomicAdd(&hist[(b * 2 + 1) * 256 + tid], h1);
}

// scan histogram, pick byte, update prefix/krem; re-zero hist; final pass sets thr
__global__ void k_scan(unsigned int* hist, unsigned int* prefix,
                       unsigned int* krem, const unsigned int* active,
                       unsigned int* thr, int pass) {
  __shared__ unsigned int lh[256];
  int s = blockIdx.x;          // 0..15 = b*2 + set
  int tid = threadIdx.x;
  lh[tid] = hist[s * 256 + tid];
  __syncthreads();
  hist[s * 256 + tid] = 0u;    // ready for next pass
  if (tid == 0 && active[s]) {
    unsigned k = krem[s];
    unsigned cum = 0; int v = 0;
    for (; v < 256; ++v) {
      unsigned c = lh[v];
      if (cum + c > k) break;
      cum += c;
    }
    if (v > 255) v = 255;
    int shift = 24 - 8 * pass;
    unsigned p = prefix[s] | ((unsigned)v << shift);
    prefix[s] = p;
    krem[s] = k - cum;
    if (pass == 3) thr[s] = p;
  }
}

// final: apply thresholds, targets, weights, inside masking; write all outputs
__global__ void k_final(const float* __restrict__ anchors,
                        const float* __restrict__ gt,
                        const float* __restrict__ im_info,
                        const float* __restrict__ prio,
                        const signed char* __restrict__ labels_ws,
                        const unsigned char* __restrict__ argmax_ws,
                        const unsigned int* __restrict__ active,
                        const unsigned int* __restrict__ thr,
                        const float* __restrict__ uw_p,
                        float* __restrict__ out) {
  __shared__ float sgt[G_N * GTS];
  int b = blockIdx.y;
  int tid = threadIdx.x;
  stage_dwords(gt + (size_t)b * G_N * GTS, sgt, tid, G_N * GTS);
  __syncthreads();

  int a = blockIdx.x * 256 + tid;
  if (a >= A_N) return;
  size_t ba = (size_t)b * A_N + a;

  float im0 = im_info[0], im1 = im_info[1];
  float a0, a1, a2, a3, a4, a5; bool inside;
  load_anc(anchors, a, im0, im1, a0, a1, a2, a3, a4, a5, inside);

  int lab = labels_ws[ba];
  unsigned key = fenc(prio[ba]);
  if (lab == 1 && active[b * 2] && key > thr[b * 2]) lab = -1;
  if (lab == 0 && active[b * 2 + 1] && key > thr[b * 2 + 1]) lab = -1;

  float t0 = 0.f, t1 = 0.f, t2 = 0.f, t3 = 0.f;
  if (inside) {
    const float* m = &sgt[(int)argmax_ws[ba] * GTS];
    float ew = a3 - a0 + 1.f, eh = a4 - a1 + 1.f;
    float ecx = a0 + 0.5f * ew, ecy = a1 + 0.5f * eh;
    float gw = m[3] - m[0] + 1.f, gh = m[4] - m[1] + 1.f;
    float gcx = m[0] + 0.5f * gw, gcy = m[1] + 0.5f * gh;
    t0 = (gcx - ecx) / ew;
    t1 = (gcy - ecy) / eh;
    t2 = logf(gw / ew);
    t3 = logf(gh / eh);
  } else {
    lab = -1;
  }
  float biw = (lab == 1) ? 1.f : 0.f;
  float bow = (lab == 0 || lab == 1) ? uw_p[0] : 0.f;

  const size_t BA = (size_t)B_N * A_N;
  out[ba] = (float)lab;
  float* tg = out + BA + ba * 4;
  tg[0] = t0; tg[1] = t1; tg[2] = t2; tg[3] = t3;
  out[BA * 5 + ba] = biw;
  out[BA * 6 + ba] = bow;
}

// ---------------- launch ----------------
extern "C" void kernel_launch(void* const* d_in, const int* in_sizes, int n_in,
                              void* d_out, int out_size, void* d_ws, size_t ws_size,
                              hipStream_t stream) {
  const float* anchors = (const float*)d_in[0];
  const float* gt      = (const float*)d_in[1];
  const float* im_info = (const float*)d_in[2];
  const float* prio    = (const float*)d_in[3];
  float* out = (float*)d_out;
  char* ws = (char*)d_ws;

  unsigned char* argmax_ws = (unsigned char*)(ws + WS_ARGMAX);
  signed char* labels_ws   = (signed char*)(ws + WS_LABELS);
  unsigned int* gtenc      = (unsigned int*)(ws + WS_GTENC);
  float* gtadj             = (float*)(ws + WS_GTADJ);
  unsigned int* cnt_fg     = (unsigned int*)(ws + WS_CNTFG);
  unsigned int* cnt_bg     = (unsigned int*)(ws + WS_CNTBG);
  unsigned int* prefix     = (unsigned int*)(ws + WS_PREFIX);
  unsigned int* krem       = (unsigned int*)(ws + WS_KREM);
  unsigned int* active     = (unsigned int*)(ws + WS_ACTIVE);
  unsigned int* thr        = (unsigned int*)(ws + WS_THR);
  float* uw                = (float*)(ws + WS_UW);
  unsigned int* hist       = (unsigned int*)(ws + WS_HIST);

  dim3 blkBA((A_N + 255) / 256, B_N);

  k_init<<<16, 256, 0, stream>>>(gtenc, cnt_fg, cnt_bg, hist);
  k_overlap<<<blkBA, 256, 0, stream>>>(anchors, gt, im_info, argmax_ws, gtenc);
  k_decode<<<1, 128, 0, stream>>>(gtenc, gtadj);
  k_labels<<<blkBA, 256, 0, stream>>>(anchors, gt, im_info, gtadj, labels_ws,
                                      cnt_fg, cnt_bg);
  k_setup<<<1, 32, 0, stream>>>(cnt_fg, cnt_bg, prefix, krem, active, thr, uw);
  dim3 blkH((A_N + 256 * HUNROLL - 1) / (256 * HUNROLL), B_N);
  for (int p = 0; p < 4; ++p) {
    k_hist<<<blkH, 256, 0, stream>>>(prio, labels_ws, prefix, active, hist, p);
    k_scan<<<16, 256, 0, stream>>>(hist, prefix, krem, active, thr, p);
  }
  k_final<<<blkBA, 256, 0, stream>>>(anchors, gt, im_info, prio, labels_ws,
                                     argmax_ws, active, thr, uw, out);
}